// FastGTConv_60309930770876
// MI455X (gfx1250) — compile-verified
//
#include <hip/hip_runtime.h>

typedef __attribute__((ext_vector_type(2))) float v2f;
typedef __attribute__((ext_vector_type(8))) float v8f;

#define T_EDGE     4
#define C_CH       2
#define N_NODES    8192
#define E_PER_TYPE 262144

// ---------------------------------------------------------------------------
// Kernel 1: zero-fill the 512 MiB output with 128-bit stores (global_store_b128)
// ---------------------------------------------------------------------------
__global__ __launch_bounds__(256) void fgt_zero_kernel(float4* __restrict__ out,
                                                       long long n4) {
  long long i = (long long)blockIdx.x * blockDim.x + threadIdx.x;
  if (i < n4) out[i] = make_float4(0.f, 0.f, 0.f, 0.f);
}

// ---------------------------------------------------------------------------
// Kernel 2: edge scatter. Each wave handles a 16-edge tile of one edge type.
//   D(16x16,f32) = A(16x4: softmax filter, rows c=0..1) x B(4x16: vals on row t)
// via v_wmma_f32_16x16x4_f32 (exact: zero-padded K contributes exactly 0).
// Lanes 0-15 then atomically add D row0 (c=0) and row1 (c=1) into out.
// ---------------------------------------------------------------------------
__global__ __launch_bounds__(256) void fgt_scatter_kernel(
    const int*   __restrict__ edge_index,   // [T, 2, E]
    const float* __restrict__ edge_values,  // [T, E]
    const float* __restrict__ weight,       // [C, T]
    float*       __restrict__ out)          // [C, N, N]
{
  const int t    = blockIdx.y;          // edge type for this block (uniform)
  const int tid  = threadIdx.x;
  const int lane = tid & 31;
  const int wave = tid >> 5;
  const int n    = lane & 15;           // tile column == edge slot 0..15
  const int half = lane >> 4;           // selects K-group per ISA layout

  const int eid = blockIdx.x * 128 + wave * 16 + n;   // 8 waves * 16 edges

  // --- softmax filter row for this lane's A-matrix row m = n (branchless) ---
  const int m   = n;
  const int row = (m < C_CH) ? m : 0;   // avoid OOB; zeroed below for m>=2
  float w0 = weight[row * T_EDGE + 0];
  float w1 = weight[row * T_EDGE + 1];
  float w2 = weight[row * T_EDGE + 2];
  float w3 = weight[row * T_EDGE + 3];
  float mx  = fmaxf(fmaxf(w0, w1), fmaxf(w2, w3));
  float e0 = __expf(w0 - mx), e1 = __expf(w1 - mx);
  float e2 = __expf(w2 - mx), e3 = __expf(w3 - mx);
  float inv = 1.0f / (e0 + e1 + e2 + e3);
  float f0 = e0 * inv, f1 = e1 * inv, f2 = e2 * inv, f3 = e3 * inv;
  float valid = (m < C_CH) ? 1.0f : 0.0f;

  // A-matrix (16x4 f32): lanes 0-15 -> M=lane, VGPR0=K0, VGPR1=K1;
  //                      lanes 16-31 -> M=lane-16, VGPR0=K2, VGPR1=K3.
  v2f a;
  a.x = valid * ((half == 0) ? f0 : f2);
  a.y = valid * ((half == 0) ? f1 : f3);

  // B-matrix (4x16 f32): VGPR0 row K=half, VGPR1 row K=2+half, col N=lane&15.
  // All 16 edges in the tile share type t, so only row t is nonzero.
  float val = edge_values[(size_t)t * E_PER_TYPE + eid];
  v2f b;
  b.x = (half     == t) ? val : 0.0f;
  b.y = (2 + half == t) ? val : 0.0f;

  v8f c = {};
  // D[c][e] = sum_k A[c][k] * B[k][e] = filt[c][t] * val[e]  (exact in f32)
  v8f d = __builtin_amdgcn_wmma_f32_16x16x4_f32(
      /*neg_a=*/false, a, /*neg_b=*/false, b,
      /*c_mod=*/(short)0, c, /*reuse_a=*/false, /*reuse_b=*/false);

  // Scatter: lanes 0-15 hold N=0..15 of D rows M=0 (VGPR0) and M=1 (VGPR1).
  if (lane < 16) {
    int r  = edge_index[((size_t)t * 2 + 0) * E_PER_TYPE + eid];
    int cc = edge_index[((size_t)t * 2 + 1) * E_PER_TYPE + eid];
    size_t off = (size_t)r * N_NODES + cc;
    atomicAdd(out + off, d[0]);                                   // channel 0
    atomicAdd(out + (size_t)N_NODES * N_NODES + off, d[1]);       // channel 1
  }
}

// ---------------------------------------------------------------------------
extern "C" void kernel_launch(void* const* d_in, const int* in_sizes, int n_in,
                              void* d_out, int out_size, void* d_ws, size_t ws_size,
                              hipStream_t stream) {
  const int*   edge_index  = (const int*)d_in[0];   // [T, 2, E] int
  const float* edge_values = (const float*)d_in[1]; // [T, E]
  const float* weight      = (const float*)d_in[2]; // [C, T]
  // d_in[3] = num_nodes scalar (compile-time constant 8192 here)
  float* out = (float*)d_out;                       // [C, N, N] float32

  // 1) zero the output (512 MiB) with b128 stores
  long long n4 = (long long)out_size / 4;
  unsigned zblocks = (unsigned)((n4 + 255) / 256);
  fgt_zero_kernel<<<zblocks, 256, 0, stream>>>((float4*)out, n4);

  // 2) WMMA-scaled edge scatter: grid.x covers E/(8 waves * 16 edges), grid.y = T
  dim3 grid(E_PER_TYPE / 128, T_EDGE);
  fgt_scatter_kernel<<<grid, 256, 0, stream>>>(edge_index, edge_values, weight, out);
}